// SpiralConv_63711544868969
// MI455X (gfx1250) — compile-verified
//
#include <hip/hip_runtime.h>
#include <cstdint>

#define BATCH 32
#define NPTS  5023
#define FEAT  64
#define SPIR  9
#define OUTC  128
#define KDIM  (FEAT * SPIR)   // 576
#define MT    64              // output rows per block (4 x 16-row WMMA tiles)
#define RT    4               // row-tiles (accumulators) per wave
#define LDSW  580             // 576 + 4 floats pad -> conflict-free A-frag loads

typedef float v2f __attribute__((ext_vector_type(2)));
typedef float v8f __attribute__((ext_vector_type(8)));

__global__ __launch_bounds__(256) void spiralconv_wmma_f32(
    const float* __restrict__ x,     // [B, N, F]
    const int*   __restrict__ adj,   // [B, N, S]
    const float* __restrict__ W,     // [OUTC, KDIM] row-major
    const float* __restrict__ bias,  // [OUTC]
    float*       __restrict__ out)   // [B, N, OUTC]
{
    extern __shared__ float sp[];            // [MT][LDSW] gathered spirals
    const int b    = blockIdx.y;
    const int base = blockIdx.x * MT;
    const int tid  = threadIdx.x;
    const size_t xb = (size_t)b * NPTS;
    const float* xbp  = x   + xb * FEAT;     // x[b]
    const int*   adjb = adj + xb * SPIR;     // adj[b]

    // -------- gather: 64 rows x 9 slots x 64 floats, async global -> LDS -----
    // segment = (m, s): one 256B feature row; 16 consecutive lanes move 16B each
    // (coalesced). Data bypasses VGPRs entirely (ASYNCcnt-tracked DMA-to-LDS).
    {
        const int f4 = tid & 15;             // float4 slot within feature row
        int seg = tid >> 4;                  // segment id 0..15 for iteration 0
        int m = seg / SPIR;                  // seg < 16 -> cheap
        int s = seg - m * SPIR;
        #pragma unroll 4
        for (int it = 0; it < (MT * SPIR) / 16; ++it) {    // 36 iterations
            int n = base + m;
            if (n > NPTS - 1) n = NPTS - 1;  // clamp phantom rows (stores masked)
            const int idx = adjb[n * SPIR + s];
            const unsigned long long g =
                (unsigned long long)(uintptr_t)(xbp + (size_t)idx * FEAT + f4 * 4);
            const uint32_t dsoff = (uint32_t)((m * LDSW + s * FEAT + f4 * 4) * 4);
            asm volatile("global_load_async_to_lds_b128 %0, %1, off"
                         :: "v"(dsoff), "v"(g) : "memory");
            // advance segment by 16: (m,s) walker, 16 = 9 + 7
            s += 7; m += 1;
            if (s >= SPIR) { s -= SPIR; m += 1; }
        }
    }
    asm volatile("s_wait_asynccnt 0x0" ::: "memory");   // my async copies done
    __syncthreads();                                    // publish LDS to all waves

    // ------------- WMMA stage: D(64x16) = A(64x576) x B(576x16) -------------
    const int lane = tid & 31;
    const int wave = tid >> 5;
    const int half = lane >> 4;       // 0: lanes 0-15, 1: lanes 16-31
    const int li   = lane & 15;
    const int col  = wave * 16 + li;  // output channel owned by this lane
    const float bb = bias[col];

    v8f acc[RT];
    #pragma unroll
    for (int t = 0; t < RT; ++t) acc[t] = (v8f){};

    // A frag (ISA layout): lane holds row (lane&15); v0=K(2*half), v1=K(2*half+1)
    // B frag (mirrored):   lane holds col; v0=W[col][k+2*half], v1=W[col][k+2*half+1]
    const float* wrow = W + (size_t)col * KDIM + 2 * half;
    const float* ap   = sp + li * LDSW + 2 * half;

    for (int k = 0; k < KDIM; k += 4) {
        const v2f bf = *(const v2f*)(wrow + k);      // 8B/lane, reused by 4 WMMAs
        #pragma unroll
        for (int t = 0; t < RT; ++t) {
            const v2f af = *(const v2f*)(ap + t * (16 * LDSW) + k);  // ds_load_b64
            acc[t] = __builtin_amdgcn_wmma_f32_16x16x4_f32(
                         false, af, false, bf, (short)0, acc[t], false, false);
        }
    }

    // ---------------- epilogue: bias + ELU (+ last-block mask) ----------------
    // C/D layout: VGPR rr, lanes 0-15 -> M=rr, lanes 16-31 -> M=rr+8
    float* orow = out + (xb + (size_t)base) * OUTC + col;
    if (base + MT < NPTS) {
        // fast path: every row valid and none is N-1 -> unconditional NT stores
        #pragma unroll
        for (int t = 0; t < RT; ++t) {
            #pragma unroll
            for (int rr = 0; rr < 8; ++rr) {
                const int m0 = t * 16 + rr + 8 * half;
                float v = acc[t][rr] + bb;
                v = (v > 0.0f) ? v : (__expf(v) - 1.0f);
                __builtin_nontemporal_store(v, orow + (size_t)m0 * OUTC);
            }
        }
    } else {
        // last block along N: bounds check + zero the N-1 row
        #pragma unroll
        for (int t = 0; t < RT; ++t) {
            #pragma unroll
            for (int rr = 0; rr < 8; ++rr) {
                const int m0 = t * 16 + rr + 8 * half;
                const int n0 = base + m0;
                float v = acc[t][rr] + bb;
                v = (v > 0.0f) ? v : (__expf(v) - 1.0f);
                if (n0 < NPTS) {
                    v = (n0 == NPTS - 1) ? 0.0f : v;
                    __builtin_nontemporal_store(v, orow + (size_t)m0 * OUTC);
                }
            }
        }
    }
}

extern "C" void kernel_launch(void* const* d_in, const int* in_sizes, int n_in,
                              void* d_out, int out_size, void* d_ws, size_t ws_size,
                              hipStream_t stream) {
    (void)in_sizes; (void)n_in; (void)out_size; (void)d_ws; (void)ws_size;
    const float* x    = (const float*)d_in[0];
    const int*   adj  = (const int*)d_in[1];
    const float* W    = (const float*)d_in[2];
    const float* bias = (const float*)d_in[3];
    float*       out  = (float*)d_out;

    dim3 grid((NPTS + MT - 1) / MT, BATCH);   // 79 x 32 blocks
    const size_t shmem = (size_t)MT * LDSW * sizeof(float);  // 148,480 B -> 2 blocks/WGP
    spiralconv_wmma_f32<<<grid, 256, shmem, stream>>>(x, adj, W, bias, out);
}